// DispLoss_22265110462933
// MI455X (gfx1250) — compile-verified
//
#include <hip/hip_runtime.h>
#include <hip/hip_bf16.h>

#define N_CLS   1000
#define FEAT    1024
#define BATCH   32768
#define CAP     96          // max samples per class we track (mean ~32.8, 96 is >10 sigma)
#define NT      63          // ceil(1000/16) tiles per dimension
#define PADM    1024        // padded row count for P
#define TEMP_INV 10.0f      // 1/TEMP

typedef float v2f __attribute__((ext_vector_type(2)));
typedef float v8f __attribute__((ext_vector_type(8)));

// Deterministic 256-thread block sum (fixed reduction order -> bitwise reproducible).
__device__ __forceinline__ float block_reduce_sum_256(float v, float* sm) {
#pragma unroll
    for (int off = 16; off > 0; off >>= 1)
        v += __shfl_xor(v, off, 32);
    const int lane = threadIdx.x & 31;
    const int w    = threadIdx.x >> 5;
    if (lane == 0) sm[w] = v;
    __syncthreads();
    if (threadIdx.x == 0) {
        float t = sm[0];
#pragma unroll
        for (int i = 1; i < 8; ++i) t += sm[i];
        sm[0] = t;
    }
    __syncthreads();
    float total = sm[0];
    __syncthreads();   // safe reuse of sm on next call
    return total;
}

// ---------------------------------------------------------------------------
// Kernel 1: per-class ordered sample lists (stable: batch order preserved).
// One thread per class; labels staged through LDS in 32KB chunks.
// ---------------------------------------------------------------------------
__global__ void build_lists_kernel(const int* __restrict__ labels,
                                   int* __restrict__ lists,
                                   int* __restrict__ counts) {
    __shared__ int sl[8192];
    const int c = blockIdx.x * 256 + threadIdx.x;   // class id (grid: 4 blocks)
    int cnt = 0;
    for (int chunk = 0; chunk < BATCH / 8192; ++chunk) {
        __syncthreads();
        for (int t = threadIdx.x; t < 8192; t += 256)
            sl[t] = labels[chunk * 8192 + t];
        __syncthreads();
        if (c < N_CLS) {
            const int base = chunk * 8192;
            for (int j = 0; j < 8192; ++j) {
                if (sl[j] == c && cnt < CAP) {
                    lists[c * CAP + cnt] = base + j;
                    ++cnt;
                }
            }
        }
    }
    if (c < N_CLS) counts[c] = cnt;
}

// ---------------------------------------------------------------------------
// Kernel 2: per-class sequential EMA chain + renormalize each step.
// Block c (0..1023): 256 threads, each owns 4 contiguous floats of the row.
// Blocks >= 1000 zero the padding rows of Ppad.
// ---------------------------------------------------------------------------
__global__ void ema_update_kernel(const float* __restrict__ protos_in,
                                  const float* __restrict__ features,
                                  const int* __restrict__ lists,
                                  const int* __restrict__ counts,
                                  float* __restrict__ Ppad) {
    __shared__ float sm[8];
    const int c = blockIdx.x;
    const int t = threadIdx.x;

    if (c >= N_CLS) {                       // zero-fill pad rows 1000..1023
        float4 z = make_float4(0.f, 0.f, 0.f, 0.f);
        ((float4*)(Ppad + (size_t)c * FEAT))[t] = z;
        return;
    }

    float4 p = ((const float4*)(protos_in + (size_t)c * FEAT))[t];
    const int cnt = counts[c];

    for (int s = 0; s < cnt; ++s) {
        const int idx = lists[c * CAP + s];
        const float4 f = ((const float4*)(features + (size_t)idx * FEAT))[t];
        float4 v;
        v.x = p.x * 0.95f + f.x * 0.05f;
        v.y = p.y * 0.95f + f.y * 0.05f;
        v.z = p.z * 0.95f + f.z * 0.05f;
        v.w = p.w * 0.95f + f.w * 0.05f;
        const float ss = v.x * v.x + v.y * v.y + v.z * v.z + v.w * v.w;
        const float S  = block_reduce_sum_256(ss, sm);
        const float inv = 1.0f / fmaxf(sqrtf(S), 1e-12f);
        p.x = v.x * inv; p.y = v.y * inv; p.z = v.z * inv; p.w = v.w * inv;
    }
    ((float4*)(Ppad + (size_t)c * FEAT))[t] = p;
}

// ---------------------------------------------------------------------------
// Kernel 3: Gram matrix G = P * P^T via V_WMMA_F32_16X16X4_F32.
// One wave per 16x16 output tile; K-loop in steps of 4.
//   A 16x4 f32: lane l -> M = l&15; VGPR0 = K {0 | 2}, VGPR1 = K {1 | 3}
//   B  4x16 f32: symmetric (N = l&15), B rows are rows of P (B = P^T tile)
//   C/D: VGPR r -> M = r + 8*(l>=16), N = l&15
// ---------------------------------------------------------------------------
__global__ void gram_wmma_kernel(const float* __restrict__ P,
                                 float* __restrict__ G) {
    const int wid = (blockIdx.x * blockDim.x + threadIdx.x) >> 5;
    if (wid >= NT * NT) return;             // uniform per wave; EXEC all-1 where active

    const int I = wid / NT;
    const int J = wid % NT;
    const int lane = threadIdx.x & 31;
    const int n    = lane & 15;
    const int koff = (lane >> 4) << 1;      // 0 for lanes 0-15, 2 for lanes 16-31

    const float* pA = P + (size_t)(I * 16 + n) * FEAT + koff;
    const float* pB = P + (size_t)(J * 16 + n) * FEAT + koff;

    v8f acc = {};
#pragma unroll 8
    for (int kk = 0; kk < FEAT; kk += 4) {
        v2f a = *(const v2f*)(pA + kk);     // 8B aligned
        v2f b = *(const v2f*)(pB + kk);
        acc = __builtin_amdgcn_wmma_f32_16x16x4_f32(
            /*neg_a=*/false, a, /*neg_b=*/false, b,
            /*c_mod=*/(short)0, acc, /*reuse_a=*/false, /*reuse_b=*/false);
    }

    const int mbase = (lane >> 4) << 3;     // 0 or 8
#pragma unroll
    for (int r = 0; r < 8; ++r) {
        const int row = I * 16 + mbase + r;
        const int col = J * 16 + n;
        G[(size_t)row * PADM + col] = acc[r];
    }
}

// ---------------------------------------------------------------------------
// Kernel 4: per-row log(sum_{j!=i} exp(10*G[i][j]) / 999)
// ---------------------------------------------------------------------------
__global__ void row_reduce_kernel(const float* __restrict__ G,
                                  float* __restrict__ rowvals) {
    __shared__ float sm[8];
    const int i = blockIdx.x;
    float sum = 0.f;
    for (int j = threadIdx.x; j < N_CLS; j += 256)
        if (j != i)
            sum += expf(G[(size_t)i * PADM + j] * TEMP_INV);
    const float S = block_reduce_sum_256(sum, sm);
    if (threadIdx.x == 0)
        rowvals[i] = logf(S * (1.0f / (float)(N_CLS - 1)));
}

// ---------------------------------------------------------------------------
// Kernel 5: loss = (TEMP/BASE_TEMP) * mean(rowvals) ; TEMP/BASE_TEMP == 1.0
// ---------------------------------------------------------------------------
__global__ void final_reduce_kernel(const float* __restrict__ rowvals,
                                    float* __restrict__ out) {
    __shared__ float sm[8];
    float sum = 0.f;
    for (int j = threadIdx.x; j < N_CLS; j += 256)
        sum += rowvals[j];
    const float S = block_reduce_sum_256(sum, sm);
    if (threadIdx.x == 0)
        out[0] = S * (1.0f / (float)N_CLS);   // * (TEMP/BASE_TEMP) == 1.0
}

// ---------------------------------------------------------------------------
extern "C" void kernel_launch(void* const* d_in, const int* in_sizes, int n_in,
                              void* d_out, int out_size, void* d_ws, size_t ws_size,
                              hipStream_t stream) {
    const float* features = (const float*)d_in[0];   // [32768,1024] f32
    const float* protos   = (const float*)d_in[1];   // [1000,1024]  f32 (unit rows)
    const int*   labels   = (const int*)d_in[2];     // [32768]      i32
    float*       out      = (float*)d_out;           // scalar f32

    // Workspace layout (~8.8 MB total)
    float* Ppad    = (float*)d_ws;                       // 1024*1024 f32  (4 MB)
    float* G       = Ppad + (size_t)PADM * FEAT;         // 1024*1024 f32  (4 MB, rows<=1008 used)
    int*   lists   = (int*)(G + (size_t)PADM * PADM);    // 1000*96 i32
    int*   counts  = lists + N_CLS * CAP;                // 1000 i32
    float* rowvals = (float*)(counts + N_CLS);           // 1000 f32

    // 1. stable per-class sample index lists
    build_lists_kernel<<<4, 256, 0, stream>>>(labels, lists, counts);

    // 2. independent per-class EMA chains (+ zero pad rows)
    ema_update_kernel<<<PADM, 256, 0, stream>>>(protos, features, lists, counts, Ppad);

    // 3. fp32 WMMA Gram matrix: 63x63 tiles, one wave each, 4 waves/block
    const int nwaves = NT * NT;
    gram_wmma_kernel<<<(nwaves + 3) / 4, 128, 0, stream>>>(Ppad, G);

    // 4. per-row masked log-mean-exp
    row_reduce_kernel<<<N_CLS, 256, 0, stream>>>(G, rowvals);

    // 5. final mean -> loss
    final_reduce_kernel<<<1, 256, 0, stream>>>(rowvals, out);
}